// HashEmbeddingLayer_30855045054746
// MI455X (gfx1250) — compile-verified
//
#include <hip/hip_runtime.h>
#include <hip/hip_bf16.h>
#include <stdint.h>

// HashEmbeddingLayer for MI455X (gfx1250), Tensor-Data-Mover gather version.
//
// Memory-bound gather-reduce (~256MB gather reads + 64MB NT writes -> ~14us
// roofline at 23.3 TB/s). The TDM gather mode (CDNA5 ISA)
// is purpose-built for this: one TENSOR_LOAD_TO_LDS with 8 x 32-bit row
// indices DMAs 8 weight rows (2 tokens x 4 hashes x 2KB = 16KB) into LDS
// without consuming wave VMEM slots or VGPRs; waves then sign-combine from
// LDS and stream the result out with non-temporal b128 stores.

typedef float    v4f   __attribute__((ext_vector_type(4)));
typedef unsigned u32x4 __attribute__((ext_vector_type(4)));
typedef int      i32x4 __attribute__((ext_vector_type(4)));
typedef int      i32x8 __attribute__((ext_vector_type(8)));

#define HE_HIDDEN      512
#define HE_NUM_HASH    4
#define HE_TPB         2                  // tokens per block (8 gather rows = 1 TDM op)
#define HE_THREADS     128                // 4 waves
#define HE_BUCKET      262144u
#define HE_BUCKET_MASK 0x3FFFFu           // BUCKET = 2^18

__global__ __launch_bounds__(HE_THREADS) void hash_embed_tdm_kernel(
    const int* __restrict__ input_ids,     // (B*T) token ids (< 128000)
    const float* __restrict__ weight,      // (262144, 512) f32, row = 2KB
    const int* __restrict__ hash_a,
    const int* __restrict__ hash_b,
    const int* __restrict__ sign_a,
    const int* __restrict__ sign_b,
    float* __restrict__ out,               // (B*T, 512) f32
    int n_tokens)
{
    __shared__ float lds_buf[HE_TPB * HE_NUM_HASH * HE_HIDDEN];  // 8 rows x 2KB = 16KB

    const int tok0 = (int)blockIdx.x * HE_TPB;
    if (tok0 >= n_tokens) return;

    // ---- Uniform hash math (SALU). Moduli are 2^18 and 2, so only the low
    // bits of the 32-bit products matter: no 64-bit arithmetic needed.
    unsigned a[HE_NUM_HASH], b[HE_NUM_HASH], sa[HE_NUM_HASH], sb[HE_NUM_HASH];
#pragma unroll
    for (int h = 0; h < HE_NUM_HASH; ++h) {
        a[h]  = (unsigned)hash_a[h];
        b[h]  = (unsigned)hash_b[h];
        sa[h] = (unsigned)sign_a[h];
        sb[h] = (unsigned)sign_b[h];
    }

    int      bucket[HE_TPB][HE_NUM_HASH];
    float    sgn[HE_TPB][HE_NUM_HASH];     // sign folded with the 1/4 mean factor
#pragma unroll
    for (int t = 0; t < HE_TPB; ++t) {
        int tid = tok0 + t;
        unsigned id = (unsigned)input_ids[(tid < n_tokens) ? tid : (n_tokens - 1)];
#pragma unroll
        for (int h = 0; h < HE_NUM_HASH; ++h) {
            bucket[t][h] = (int)((id * a[h] + b[h]) & HE_BUCKET_MASK);
            sgn[t][h]    = ((id * sa[h] + sb[h]) & 1u) ? 0.25f : -0.25f;
        }
    }

    // ---- One wave builds the TDM gather descriptor (all fields wave-uniform
    // -> SGPRs) and issues TENSOR_LOAD_TO_LDS. TDM ignores EXEC, so gate with
    // a scalar (readfirstlane-derived) branch so waves 1..3 skip the issue.
    const unsigned wave_id = __builtin_amdgcn_readfirstlane((unsigned)threadIdx.x >> 5);
    if (wave_id == 0) {
        const unsigned lds_off = (unsigned)(uintptr_t)&lds_buf[0];          // LDS byte offset
        const unsigned long long gaddr = (unsigned long long)(uintptr_t)weight;

        // D# group 0 (ISA §8.3): count=1 | gather_index_size=32b | gather_mode=1,
        // lds_addr, global_addr[56:0], type=2 ("image").
        u32x4 g0;
        g0[0] = 0xC0000001u;
        g0[1] = lds_off;
        g0[2] = (unsigned)(gaddr & 0xFFFFFFFFu);
        g0[3] = (unsigned)((gaddr >> 32) & 0x01FFFFFFu) | (2u << 30);

        // D# group 1 (ISA §8.4): data_size=4B; tensor_dim0=512 (row length),
        // tensor_dim1=262144 (row count, OOB bound); tile_dim0=512,
        // tile_dim1=8 valid indices; tensor_dim0_stride=512 elements.
        i32x8 g1;
        g1[0] = 0x00020000;                         // data_size=2 -> 4 bytes
        g1[1] = (int)(HE_HIDDEN << 16);             // tensor_dim0[15:0] at bit 48
        g1[2] = 0;                                  // tensor_dim0 hi / tensor_dim1 lo16 (=0)
        g1[3] = (int)(((unsigned)HE_HIDDEN << 16) | (HE_BUCKET >> 16));  // tile_dim0 | tensor_dim1 hi
        g1[4] = HE_TPB * HE_NUM_HASH;               // tile_dim1 = 8 gather indices
        g1[5] = HE_HIDDEN;                          // tensor_dim0_stride lo
        g1[6] = 0;
        g1[7] = 0;

        // D# groups 2/3 (gather mode, 32-bit indices): row_index_0..7.
        i32x4 g2, g3;
#pragma unroll
        for (int h = 0; h < HE_NUM_HASH; ++h) {
            g2[h] = bucket[0][h];
            g3[h] = bucket[1][h];
        }

        asm volatile("tensor_load_to_lds %0, %1, %2, %3"
                     :
                     : "s"(g0), "s"(g1), "s"(g2), "s"(g3)
                     : "memory");
        __builtin_amdgcn_s_wait_tensorcnt(0);       // gather rows resident in LDS
    }
    __syncthreads();                                // publish LDS to all 4 waves

    // ---- Combine from LDS (conflict-free ds_load_b128: each lane reads two adjacent
    // 16B chunks of a 2KB row) and stream out with non-temporal b128 stores.
    const int tok = (int)threadIdx.x >> 6;          // 0 or 1
    const int col = ((int)threadIdx.x & 63) * 8;    // 8 floats per thread
    const float* base = lds_buf + tok * (HE_NUM_HASH * HE_HIDDEN) + col;

    float s[HE_NUM_HASH];
#pragma unroll
    for (int h = 0; h < HE_NUM_HASH; ++h)
        s[h] = (tok == 0) ? sgn[0][h] : sgn[1][h];

    v4f acc0 = {0.f, 0.f, 0.f, 0.f};
    v4f acc1 = {0.f, 0.f, 0.f, 0.f};
#pragma unroll
    for (int h = 0; h < HE_NUM_HASH; ++h) {
        v4f x0 = *(const v4f*)(base + h * HE_HIDDEN);
        v4f x1 = *(const v4f*)(base + h * HE_HIDDEN + 4);
#pragma unroll
        for (int c = 0; c < 4; ++c) {
            acc0[c] = __builtin_fmaf(s[h], x0[c], acc0[c]);
            acc1[c] = __builtin_fmaf(s[h], x1[c], acc1[c]);
        }
    }

    const int tid = tok0 + tok;
    if (tid < n_tokens) {
        float* dst = out + (size_t)tid * HE_HIDDEN + col;
        __builtin_nontemporal_store(acc0, (v4f*)dst);
        __builtin_nontemporal_store(acc1, (v4f*)(dst + 4));
    }
}

extern "C" void kernel_launch(void* const* d_in, const int* in_sizes, int n_in,
                              void* d_out, int out_size, void* d_ws, size_t ws_size,
                              hipStream_t stream) {
    const int*   input_ids = (const int*)d_in[0];
    const float* weight    = (const float*)d_in[1];
    const int*   hash_a    = (const int*)d_in[2];
    const int*   hash_b    = (const int*)d_in[3];
    const int*   sign_a    = (const int*)d_in[4];
    const int*   sign_b    = (const int*)d_in[5];
    float* out = (float*)d_out;

    const int n_tokens = in_sizes[0];  // B*T = 32768
    const int n_blocks = (n_tokens + HE_TPB - 1) / HE_TPB;

    hash_embed_tdm_kernel<<<n_blocks, HE_THREADS, 0, stream>>>(
        input_ids, weight, hash_a, hash_b, sign_a, sign_b, out, n_tokens);
}